// HAN_56152402427952
// MI455X (gfx1250) — compile-verified
//
#include <hip/hip_runtime.h>
#include <hip/hip_bf16.h>
#include <math.h>

// ---------------------------------------------------------------------------
// HAN forward for MI455X (gfx1250, wave32). fp32 WMMA (16x16x4) for GEMMs.
// ---------------------------------------------------------------------------

#define N_AUTHOR 50000
#define N_PAPER  50000
#define NEDGE    400000
#define HID      256
#define NHEAD    32
#define DHEAD    8
#define C_OUT    16

typedef float v2f __attribute__((ext_vector_type(2)));
typedef float v8f __attribute__((ext_vector_type(8)));

#define WMMA_F32(a, b, c) \
  __builtin_amdgcn_wmma_f32_16x16x4_f32(false, (a), false, (b), (short)0, (c), false, false)

// -------- order-preserving float<->uint encoding for atomicMax --------------
__device__ __forceinline__ unsigned fenc(float f) {
  unsigned u = __float_as_uint(f);
  return (u & 0x80000000u) ? ~u : (u | 0x80000000u);
}
__device__ __forceinline__ float fdec(unsigned k) {
  return (k & 0x80000000u) ? __uint_as_float(k & 0x7FFFFFFFu)
                           : __uint_as_float(~k);
}

// ---------------------------------------------------------------------------
// GEMM: Out[panel*16 .. +16, :256] = A_panel(16x256) @ W(256x256) + bias
// grid = (M/16, 2), block = 256 (8 waves). wave w of half blockIdx.y covers
// N-tile (blockIdx.y*8 + w). fp32 WMMA 16x16x4, K loop of 64 steps.
// ---------------------------------------------------------------------------
__global__ __launch_bounds__(256) void gemm256_bias(
    const float* __restrict__ A, const float* __restrict__ W,
    const float* __restrict__ bias, float* __restrict__ Out) {
  __shared__ float As[16 * 256];
  const int panel = blockIdx.x;
  const int t = threadIdx.x;
  const float* Ap = A + (size_t)panel * 16 * 256;
#pragma unroll
  for (int i = 0; i < 16; ++i) As[i * 256 + t] = Ap[i * 256 + t];
  __syncthreads();

  const int wave = t >> 5, lane = t & 31;
  const int col = (blockIdx.y * 8 + wave) * 16 + (lane & 15);
  const int row = lane & 15;
  const int khalf = (lane >> 4) << 1;  // 0 or 2

  v8f c = {};
  for (int k = 0; k < 256; k += 4) {
    const int kk = k + khalf;
    v2f a, b;
    a.x = As[row * 256 + kk];
    a.y = As[row * 256 + kk + 1];
    b.x = W[kk * 256 + col];
    b.y = W[(kk + 1) * 256 + col];
    c = WMMA_F32(a, b, c);
  }
  const float bv = bias[col];
  const int mbase = (lane < 16) ? 0 : 8;
#pragma unroll
  for (int r = 0; r < 8; ++r)
    Out[(size_t)(panel * 16 + mbase + r) * 256 + col] = c[r] + bv;
}

// ---------------------------------------------------------------------------
// a[n,h] = sum_d h[n, h*8+d] * att[h*8+d]
// ---------------------------------------------------------------------------
__global__ void head_dot(const float* __restrict__ h,
                         const float* __restrict__ att,
                         float* __restrict__ a, int N) {
  int idx = blockIdx.x * blockDim.x + threadIdx.x;
  if (idx >= N * NHEAD) return;
  int n = idx >> 5, hh = idx & 31;
  const float* hp = h + (size_t)n * HID + hh * DHEAD;
  const float* ap = att + hh * DHEAD;
  float s = 0.f;
#pragma unroll
  for (int d = 0; d < DHEAD; ++d) s += hp[d] * ap[d];
  a[idx] = s;
}

__device__ __forceinline__ float edge_logit(const float* aS, const float* aD,
                                            int s, int d, int hh) {
  float v = aS[s * NHEAD + hh] + aD[d * NHEAD + hh];
  return v > 0.f ? v : 0.2f * v;  // leaky_relu
}

// -------- pass 1: segment max (ordered-uint atomicMax) ----------------------
__global__ void edge_max(const int* __restrict__ src, const int* __restrict__ dst,
                         const float* __restrict__ aS, const float* __restrict__ aD,
                         unsigned* __restrict__ amax) {
  int idx = blockIdx.x * blockDim.x + threadIdx.x;
  if (idx >= NEDGE * NHEAD) return;
  int e = idx >> 5, hh = idx & 31;
  int s = src[e], d = dst[e];
  float v = edge_logit(aS, aD, s, d, hh);
  atomicMax(&amax[d * NHEAD + hh], fenc(v));
}

// -------- pass 2: denom = segment sum of exp(v - max) -----------------------
__global__ void edge_expsum(const int* __restrict__ src, const int* __restrict__ dst,
                            const float* __restrict__ aS, const float* __restrict__ aD,
                            const unsigned* __restrict__ amax,
                            float* __restrict__ denom) {
  int idx = blockIdx.x * blockDim.x + threadIdx.x;
  if (idx >= NEDGE * NHEAD) return;
  int e = idx >> 5, hh = idx & 31;
  int s = src[e], d = dst[e];
  float v = edge_logit(aS, aD, s, d, hh);
  float m = fdec(amax[d * NHEAD + hh]);
  atomicAdd(&denom[d * NHEAD + hh], expf(v - m));
}

// -------- pass 3: out[dst] += h_src[src] * alpha ----------------------------
__global__ void edge_scatter(const int* __restrict__ src, const int* __restrict__ dst,
                             const float* __restrict__ aS, const float* __restrict__ aD,
                             const unsigned* __restrict__ amax,
                             const float* __restrict__ denom,
                             const float* __restrict__ hsrc,
                             float* __restrict__ out) {
  int idx = blockIdx.x * blockDim.x + threadIdx.x;
  if (idx >= NEDGE * NHEAD) return;
  int e = idx >> 5, hh = idx & 31;
  int s = src[e], d = dst[e];
  float v = edge_logit(aS, aD, s, d, hh);
  float m = fdec(amax[d * NHEAD + hh]);
  float alpha = expf(v - m) / denom[d * NHEAD + hh];
  const float* hp = hsrc + (size_t)s * HID + hh * DHEAD;
  float* op = out + (size_t)d * HID + hh * DHEAD;
#pragma unroll
  for (int dd = 0; dd < DHEAD; ++dd) atomicAdd(&op[dd], hp[dd] * alpha);
}

__global__ void relu_k(float* __restrict__ x, int n) {
  int i = blockIdx.x * blockDim.x + threadIdx.x;
  if (i < n) {
    float v = x[i];
    x[i] = v > 0.f ? v : 0.f;
  }
}

// ---------------------------------------------------------------------------
// score[t] += sum over this panel of q . tanh(out_t @ Wk + bk)
// grid = (M/16, 2 types), block = 256; each wave owns 2 N-tiles.
// ---------------------------------------------------------------------------
__global__ __launch_bounds__(256) void semantic_score(
    const float* __restrict__ outW, const float* __restrict__ outC,
    const float* __restrict__ Wk, const float* __restrict__ bk,
    const float* __restrict__ q, float* __restrict__ score) {
  __shared__ float As[16 * 256];
  __shared__ float red[256];
  const float* A = (blockIdx.y == 0) ? outW : outC;
  const int panel = blockIdx.x;
  const int t = threadIdx.x;
  const float* Ap = A + (size_t)panel * 16 * 256;
#pragma unroll
  for (int i = 0; i < 16; ++i) As[i * 256 + t] = Ap[i * 256 + t];
  __syncthreads();

  const int wave = t >> 5, lane = t & 31;
  const int row = lane & 15;
  const int khalf = (lane >> 4) << 1;

  float partial = 0.f;
  for (int ct = 0; ct < 2; ++ct) {
    const int col = (wave * 2 + ct) * 16 + (lane & 15);
    v8f c = {};
    for (int k = 0; k < 256; k += 4) {
      const int kk = k + khalf;
      v2f a, b;
      a.x = As[row * 256 + kk];
      a.y = As[row * 256 + kk + 1];
      b.x = Wk[kk * 256 + col];
      b.y = Wk[(kk + 1) * 256 + col];
      c = WMMA_F32(a, b, c);
    }
    const float qv = q[col], bb = bk[col];
#pragma unroll
    for (int r = 0; r < 8; ++r) partial += qv * tanhf(c[r] + bb);
  }
  red[t] = partial;
  __syncthreads();
  for (int s = 128; s > 0; s >>= 1) {
    if (t < s) red[t] += red[t + s];
    __syncthreads();
  }
  if (t == 0) atomicAdd(&score[blockIdx.y], red[0]);
}

__global__ void softmax2(const float* __restrict__ score, float* __restrict__ attn) {
  float s0 = score[0] / (float)N_PAPER;
  float s1 = score[1] / (float)N_PAPER;
  float m = fmaxf(s0, s1);
  float e0 = expf(s0 - m), e1 = expf(s1 - m);
  float inv = 1.f / (e0 + e1);
  attn[0] = e0 * inv;
  attn[1] = e1 * inv;
}

// ---------------------------------------------------------------------------
// O = (attn0*outW + attn1*outC) @ W_out + b_out   (N = 16 -> one tile/wave)
// grid = M/16, block = 32 (one wave).
// ---------------------------------------------------------------------------
__global__ __launch_bounds__(32) void final_out(
    const float* __restrict__ outW, const float* __restrict__ outC,
    const float* __restrict__ attn, const float* __restrict__ Wout,
    const float* __restrict__ bout, float* __restrict__ O) {
  const int panel = blockIdx.x;
  const int lane = threadIdx.x;
  const float a0 = attn[0], a1 = attn[1];
  const int row = lane & 15;
  const int khalf = (lane >> 4) << 1;
  const int col = lane & 15;
  const float* Aw = outW + (size_t)panel * 16 * 256;
  const float* Ac = outC + (size_t)panel * 16 * 256;

  v8f c = {};
  for (int k = 0; k < 256; k += 4) {
    const int kk = k + khalf;
    v2f a, b;
    a.x = a0 * Aw[row * 256 + kk] + a1 * Ac[row * 256 + kk];
    a.y = a0 * Aw[row * 256 + kk + 1] + a1 * Ac[row * 256 + kk + 1];
    b.x = Wout[kk * C_OUT + col];
    b.y = Wout[(kk + 1) * C_OUT + col];
    c = WMMA_F32(a, b, c);
  }
  const float bv = bout[col];
  const int mbase = (lane < 16) ? 0 : 8;
#pragma unroll
  for (int r = 0; r < 8; ++r)
    O[(size_t)(panel * 16 + mbase + r) * C_OUT + col] = c[r] + bv;
}

// ---------------------------------------------------------------------------
extern "C" void kernel_launch(void* const* d_in, const int* in_sizes, int n_in,
                              void* d_out, int out_size, void* d_ws, size_t ws_size,
                              hipStream_t stream) {
  (void)n_in; (void)out_size; (void)ws_size;

  const float* x_author = (const float*)d_in[0];
  const float* x_paper  = (const float*)d_in[1];
  const float* W_author = (const float*)d_in[2];
  const float* b_author = (const float*)d_in[3];
  const float* W_paper  = (const float*)d_in[4];
  const float* b_paper  = (const float*)d_in[5];
  const float* att_sw   = (const float*)d_in[6];
  const float* att_dw   = (const float*)d_in[7];
  const float* att_sc   = (const float*)d_in[8];
  const float* att_dc   = (const float*)d_in[9];
  const float* Wk       = (const float*)d_in[10];
  const float* bk       = (const float*)d_in[11];
  const float* q        = (const float*)d_in[12];
  const float* W_out    = (const float*)d_in[13];
  const float* b_out    = (const float*)d_in[14];
  const int*   e_writes = (const int*)d_in[15];
  const int*   e_cites  = (const int*)d_in[16];
  const int E = in_sizes[15] / 2;

  // ---- workspace layout (floats) ----
  const size_t NH = (size_t)N_PAPER * HID;     // 12.8M
  const size_t NA = (size_t)N_PAPER * NHEAD;   // 1.6M
  float* h_author = (float*)d_ws;
  float* h_paper  = h_author + NH;
  float* out_w    = h_paper + NH;
  float* out_c    = out_w + NH;                // adjacent to out_w for one relu
  float* a_sw     = out_c + NH;
  float* a_dw     = a_sw + NA;
  float* a_sc     = a_dw + NA;
  float* a_dc     = a_sc + NA;
  unsigned* amax_w = (unsigned*)(a_dc + NA);
  unsigned* amax_c = amax_w + NA;              // adjacent for one memset
  float* denom_w  = (float*)(amax_c + NA);
  float* denom_c  = denom_w + NA;              // adjacent for one memset
  float* score    = denom_c + NA;
  float* attn     = score + 2;

  // ---- init accumulators (graph-capture-legal memsets) ----
  hipMemsetAsync(out_w,  0, 2 * NH * sizeof(float), stream);   // out_w + out_c
  hipMemsetAsync(amax_w, 0, 2 * NA * sizeof(unsigned), stream);// key 0 < enc(any)
  hipMemsetAsync(denom_w,0, 2 * NA * sizeof(float), stream);
  hipMemsetAsync(score,  0, 2 * sizeof(float), stream);

  const int panels = N_PAPER / 16;  // 3125

  // ---- node projections (fp32 WMMA GEMM) ----
  gemm256_bias<<<dim3(panels, 2), 256, 0, stream>>>(x_author, W_author, b_author, h_author);
  gemm256_bias<<<dim3(panels, 2), 256, 0, stream>>>(x_paper,  W_paper,  b_paper,  h_paper);

  // ---- per-head attention logits ----
  const int ndot = N_PAPER * NHEAD;
  head_dot<<<(ndot + 255) / 256, 256, 0, stream>>>(h_author, att_sw, a_sw, N_AUTHOR);
  head_dot<<<(ndot + 255) / 256, 256, 0, stream>>>(h_paper,  att_dw, a_dw, N_PAPER);
  head_dot<<<(ndot + 255) / 256, 256, 0, stream>>>(h_paper,  att_sc, a_sc, N_PAPER);
  head_dot<<<(ndot + 255) / 256, 256, 0, stream>>>(h_paper,  att_dc, a_dc, N_PAPER);

  // ---- edge-type GAT: segment softmax + scatter ----
  const int* ws_src = e_writes;       const int* ws_dst = e_writes + E;
  const int* cs_src = e_cites;        const int* cs_dst = e_cites + E;
  const int nEH = E * NHEAD;
  const int egrid = (nEH + 255) / 256;

  edge_max<<<egrid, 256, 0, stream>>>(ws_src, ws_dst, a_sw, a_dw, amax_w);
  edge_max<<<egrid, 256, 0, stream>>>(cs_src, cs_dst, a_sc, a_dc, amax_c);
  edge_expsum<<<egrid, 256, 0, stream>>>(ws_src, ws_dst, a_sw, a_dw, amax_w, denom_w);
  edge_expsum<<<egrid, 256, 0, stream>>>(cs_src, cs_dst, a_sc, a_dc, amax_c, denom_c);
  edge_scatter<<<egrid, 256, 0, stream>>>(ws_src, ws_dst, a_sw, a_dw, amax_w, denom_w,
                                          h_author, out_w);
  edge_scatter<<<egrid, 256, 0, stream>>>(cs_src, cs_dst, a_sc, a_dc, amax_c, denom_c,
                                          h_paper, out_c);

  relu_k<<<(int)((2 * NH + 255) / 256), 256, 0, stream>>>(out_w, (int)(2 * NH));

  // ---- semantic attention (WMMA GEMM fused with tanh/q-dot/reduction) ----
  semantic_score<<<dim3(panels, 2), 256, 0, stream>>>(out_w, out_c, Wk, bk, q, score);
  softmax2<<<1, 1, 0, stream>>>(score, attn);

  // ---- fused output projection (WMMA) ----
  final_out<<<panels, 32, 0, stream>>>(out_w, out_c, attn, W_out, b_out, (float*)d_out);
}